// BatchNormSPDMeanScalarVariance_78134045049544
// MI455X (gfx1250) — compile-verified
//
#include <hip/hip_runtime.h>
#include <math.h>

// ---------------------------------------------------------------------------
// SPD BatchNorm (affine-invariant) for MI455X / gfx1250.
// All matrix functions (sqrtm, logm, expm, powm) are computed with
// GEMM-only iterations (Newton-Schulz / scaling-and-squaring) so that the
// entire pipeline runs on V_WMMA_F32_16X16X4_F32.
// ---------------------------------------------------------------------------

#define NF     64
#define MAT    (NF * NF)      // 4096 elements per matrix
#define BATCH  4096
#define TPB    256            // 8 wave32 waves per workgroup

#define NS_ITERS        9     // Newton-Schulz iterations per square root
#define LOG_SQRT_LEVELS 5     // inverse scaling-and-squaring depth for logm
#define LOG_TERMS       8     // log(I+E) series terms
#define EXP_K           5     // expm squaring levels
#define EXP_TERMS       10    // expm Taylor terms

typedef __attribute__((ext_vector_type(2))) float v2f;
typedef __attribute__((ext_vector_type(8))) float v8f;

// ---------------------------------------------------------------------------
// Workgroup GEMM: C = A * B, 64x64 f32 row-major, all in LDS.
// 8 waves; each wave computes two 16x16 tiles; K=64 as 16 chained
// v_wmma_f32_16x16x4_f32 ops. Fragment layouts follow CDNA5 ISA:
//   A 16x4:  lanes 0-15 rows, VGPR0/1 = K pair, lane-half selects K pair
//   C 16x16: VGPR r -> row r (+8 for lanes 16-31), col = lane & 15
// ---------------------------------------------------------------------------
__device__ __forceinline__ void gemm64(const float* __restrict__ A,
                                       const float* __restrict__ B,
                                       float* __restrict__ C) {
  __syncthreads();
  const int lane = threadIdx.x & 31;
  const int wave = threadIdx.x >> 5;
  const int half = lane >> 4;   // 0 or 1
  const int ln   = lane & 15;
  for (int t = wave * 2; t < wave * 2 + 2; ++t) {
    const int tr = (t >> 2) * 16;
    const int tc = (t & 3) * 16;
    v8f acc = {0.f, 0.f, 0.f, 0.f, 0.f, 0.f, 0.f, 0.f};
    const int arow = tr + ln;
    const int bcol = tc + ln;
#pragma unroll
    for (int k0 = 0; k0 < NF; k0 += 4) {
      const int ka = k0 + half * 2;
      v2f a, b;
      a.x = A[arow * NF + ka];
      a.y = A[arow * NF + ka + 1];
      b.x = B[ka * NF + bcol];
      b.y = B[(ka + 1) * NF + bcol];
      acc = __builtin_amdgcn_wmma_f32_16x16x4_f32(
          /*neg_a=*/false, a, /*neg_b=*/false, b,
          /*c_mod=*/(short)0, acc, /*reuse_a=*/false, /*reuse_b=*/false);
    }
    const int rbase = tr + half * 8;
#pragma unroll
    for (int r = 0; r < 8; ++r)
      C[(rbase + r) * NF + tc + ln] = acc[r];
  }
  __syncthreads();
}

// --------------------------- elementwise helpers ---------------------------
__device__ __forceinline__ void mat_copy2(float* D, const float* S) {
  for (int i = threadIdx.x; i < MAT; i += TPB) D[i] = S[i];
  __syncthreads();
}
__device__ __forceinline__ void ew_scale_from(float* D, const float* S, float s) {
  for (int i = threadIdx.x; i < MAT; i += TPB) D[i] = S[i] * s;
  __syncthreads();
}
__device__ __forceinline__ void ew_scale(float* D, float s) {
  for (int i = threadIdx.x; i < MAT; i += TPB) D[i] *= s;
  __syncthreads();
}
__device__ __forceinline__ void ew_add(float* D, const float* S, float c) {
  for (int i = threadIdx.x; i < MAT; i += TPB) D[i] += c * S[i];
  __syncthreads();
}
__device__ __forceinline__ void ew_identity(float* D) {
  for (int i = threadIdx.x; i < MAT; i += TPB)
    D[i] = ((i >> 6) == (i & 63)) ? 1.f : 0.f;
  __syncthreads();
}
__device__ __forceinline__ void ew_w_update(float* W) {  // W = 1.5I - 0.5*W
  for (int i = threadIdx.x; i < MAT; i += TPB) {
    const float d = ((i >> 6) == (i & 63)) ? 1.5f : 0.f;
    W[i] = d - 0.5f * W[i];
  }
  __syncthreads();
}
__device__ __forceinline__ void ew_sub_identity(float* A) {
  for (int i = threadIdx.x; i < MAT; i += TPB)
    if ((i >> 6) == (i & 63)) A[i] -= 1.f;
  __syncthreads();
}
__device__ __forceinline__ void ew_add_identity(float* A) {
  for (int i = threadIdx.x; i < MAT; i += TPB)
    if ((i >> 6) == (i & 63)) A[i] += 1.f;
  __syncthreads();
}

__device__ __forceinline__ float block_reduce(float v, float* red) {
  red[threadIdx.x] = v;
  __syncthreads();
  for (int s = TPB >> 1; s > 0; s >>= 1) {
    if ((int)threadIdx.x < s) red[threadIdx.x] += red[threadIdx.x + s];
    __syncthreads();
  }
  const float r = red[0];
  __syncthreads();
  return r;
}
__device__ __forceinline__ float frob2(const float* A, float* red) {
  float s = 0.f;
  for (int i = threadIdx.x; i < MAT; i += TPB) { const float v = A[i]; s += v * v; }
  return block_reduce(s, red);
}

// ---------------------------------------------------------------------------
// Coupled Newton-Schulz: given SPD A, produce pY -> sqrt(A/c), pZ -> (A/c)^-1/2
// with c = ||A||_F (spectrum of A/c in (0,1] => convergence). 3 GEMMs / iter.
// Buffers rotate among {A,Y,Z,W}; W is also used as the reduction scratch.
// ---------------------------------------------------------------------------
__device__ __forceinline__ void ns_sqrt_pair(float* A, float* Y, float* Z, float* W,
                                             float** oY, float** oZ, float* oC) {
  const float c2 = frob2(A, W);
  const float c = sqrtf(c2) + 1e-30f;
  ew_scale_from(Y, A, 1.0f / c);
  ew_identity(Z);
  float* pY = Y; float* pZ = Z; float* pW = W; float* pT = A;
#pragma unroll 1
  for (int it = 0; it < NS_ITERS; ++it) {
    gemm64(pZ, pY, pW);      // pW = Z*Y
    ew_w_update(pW);         // pW = 1.5I - 0.5*Z*Y
    gemm64(pY, pW, pT);      // new Y
    gemm64(pW, pZ, pY);      // new Z (into old Y slot)
    float* oldY = pY; float* oldZ = pZ;
    pY = pT; pZ = oldY; pT = oldZ;   // pW stays scratch
  }
  *oY = pY; *oZ = pZ; *oC = c;
}

__device__ __forceinline__ void sqrtm_inplace(float* A, float* Y, float* Z, float* W) {
  float *pY, *pZ, c;
  ns_sqrt_pair(A, Y, Z, W, &pY, &pZ, &c);
  ew_scale_from(A, pY, sqrtf(c));   // sqrt(A) = sqrt(c) * sqrt(A/c)
}

// logm via inverse scaling-and-squaring: s repeated square roots, then
// log(I+E) Taylor series, scaled back by 2^s. Result in A.
__device__ __forceinline__ void logm_inplace(float* A, float* Y, float* Z, float* W) {
#pragma unroll 1
  for (int l = 0; l < LOG_SQRT_LEVELS; ++l) {
    float *pY, *pZ, c;
    ns_sqrt_pair(A, Y, Z, W, &pY, &pZ, &c);
    ew_scale_from(A, pY, sqrtf(c));
  }
  ew_sub_identity(A);     // A = E,  ||E|| small
  mat_copy2(Y, A);        // P = E
  mat_copy2(Z, A);        // sum = E
  float* pP = Y; float* pT = W;
  float sign = -1.f;
#pragma unroll 1
  for (int j = 2; j <= LOG_TERMS; ++j) {
    gemm64(pP, A, pT);                  // P_j = P_{j-1} * E
    float* t = pP; pP = pT; pT = t;
    ew_add(Z, pP, sign / (float)j);
    sign = -sign;
  }
  ew_scale_from(A, Z, (float)(1 << LOG_SQRT_LEVELS));
}

// expm via scaling-and-squaring + Taylor. Result in A.
__device__ __forceinline__ void expm_inplace(float* A, float* Y, float* Z, float* W) {
  ew_scale(A, 1.0f / (float)(1 << EXP_K));
  mat_copy2(Y, A);                       // term = S
  for (int i = threadIdx.x; i < MAT; i += TPB)
    Z[i] = A[i] + (((i >> 6) == (i & 63)) ? 1.f : 0.f);   // sum = I + S
  __syncthreads();
  float* pT = Y; float* pS = W;
#pragma unroll 1
  for (int j = 2; j <= EXP_TERMS; ++j) {
    gemm64(pT, A, pS);                   // term * S
    ew_scale(pS, 1.0f / (float)j);
    float* t = pT; pT = pS; pS = t;
    ew_add(Z, pT, 1.0f);
  }
  float* cur = Z; float* nxt = A;        // A free after series
#pragma unroll 1
  for (int k = 0; k < EXP_K; ++k) {
    gemm64(cur, cur, nxt);
    float* t = cur; cur = nxt; nxt = t;
  }
  if (cur != A) mat_copy2(A, cur);
}

// ------------------------------- kernels -----------------------------------

// dst = mean over batch of src[b]  (16 blocks x 256 threads, coalesced)
__global__ __launch_bounds__(TPB) void mean_batch_kernel(
    const float* __restrict__ src, float* __restrict__ dst) {
  const int i = blockIdx.x * TPB + threadIdx.x;
  if (i >= MAT) return;
  float s = 0.f;
  for (int b = 0; b < BATCH; ++b) s += src[(size_t)b * MAT + i];
  dst[i] = s * (1.0f / (float)BATCH);
}

// Gh = sqrtm(G), Gih = inv_sqrtm(G)  (one Newton-Schulz run gives both)
__global__ __launch_bounds__(TPB) void g_prep_kernel(
    const float* __restrict__ G, float* __restrict__ Gh, float* __restrict__ Gih) {
  extern __shared__ float sm[];
  float* A = sm; float* Y = sm + MAT; float* Z = sm + 2 * MAT; float* W = sm + 3 * MAT;
  for (int i = threadIdx.x; i < MAT; i += TPB) A[i] = G[i];
  __syncthreads();
  float *pY, *pZ, c;
  ns_sqrt_pair(A, Y, Z, W, &pY, &pZ, &c);
  const float sc = sqrtf(c), isc = 1.0f / sqrtf(c);
  for (int i = threadIdx.x; i < MAT; i += TPB) {
    Gh[i]  = pY[i] * sc;
    Gih[i] = pZ[i] * isc;
  }
}

// L_b = logm(Gih * X_b * Gih)
__global__ __launch_bounds__(TPB) void karcher_pass_kernel(
    const float* __restrict__ X, const float* __restrict__ Gih,
    float* __restrict__ Lout) {
  extern __shared__ float sm[];
  float* A = sm; float* Y = sm + MAT; float* Z = sm + 2 * MAT; float* W = sm + 3 * MAT;
  const size_t b = blockIdx.x;
  for (int i = threadIdx.x; i < MAT; i += TPB) { A[i] = X[b * MAT + i]; Y[i] = Gih[i]; }
  gemm64(Y, A, Z);            // Gih * X
  gemm64(Z, Y, A);            // M = Gih * X * Gih
  logm_inplace(A, Y, Z, W);
  for (int i = threadIdx.x; i < MAT; i += TPB) Lout[b * MAT + i] = A[i];
}

// G = Gh * expm(Tmean) * Gh  (symmetrized)
__global__ __launch_bounds__(TPB) void g_update_kernel(
    const float* __restrict__ Tm, const float* __restrict__ Gh,
    float* __restrict__ G) {
  extern __shared__ float sm[];
  float* A = sm; float* Y = sm + MAT; float* Z = sm + 2 * MAT; float* W = sm + 3 * MAT;
  for (int i = threadIdx.x; i < MAT; i += TPB) A[i] = Tm[i];
  __syncthreads();
  expm_inplace(A, Y, Z, W);
  for (int i = threadIdx.x; i < MAT; i += TPB) Y[i] = Gh[i];
  gemm64(Y, A, Z);            // Gh * expT
  gemm64(Z, Y, W);            // * Gh
  for (int i = threadIdx.x; i < MAT; i += TPB)
    G[i] = 0.5f * (W[i] + W[((i & 63) << 6) | (i >> 6)]);
}

// partial[b] = ||logm(Gih X_b Gih)||_F^2
__global__ __launch_bounds__(TPB) void std_pass_kernel(
    const float* __restrict__ X, const float* __restrict__ Gih,
    float* __restrict__ partial) {
  extern __shared__ float sm[];
  float* A = sm; float* Y = sm + MAT; float* Z = sm + 2 * MAT; float* W = sm + 3 * MAT;
  const size_t b = blockIdx.x;
  for (int i = threadIdx.x; i < MAT; i += TPB) { A[i] = X[b * MAT + i]; Y[i] = Gih[i]; }
  gemm64(Y, A, Z);
  gemm64(Z, Y, A);
  logm_inplace(A, Y, Z, W);
  float s = 0.f;
  for (int i = threadIdx.x; i < MAT; i += TPB) { const float v = A[i]; s += v * v; }
  const float tot = block_reduce(s, W);
  if (threadIdx.x == 0) partial[blockIdx.x] = tot;
}

// std = sqrt(mean(partial)); p = softplus(stdbias)/std
__global__ __launch_bounds__(TPB) void finalize_kernel(
    const float* __restrict__ partial, const float* __restrict__ stdbias,
    float* __restrict__ scal) {
  __shared__ float red[TPB];
  float s = 0.f;
  for (int i = threadIdx.x; i < BATCH; i += TPB) s += partial[i];
  const float tot = block_reduce(s, red);
  if (threadIdx.x == 0) {
    const float sd = sqrtf(tot / (float)BATCH);
    scal[0] = log1pf(expf(stdbias[0])) / sd;
  }
}

// Bh = sqrtm(softplus_m(covbias)) with softplus_m(C) = logm(I + expm(sym(C)))
__global__ __launch_bounds__(TPB) void bias_prep_kernel(
    const float* __restrict__ Cb, float* __restrict__ Bh) {
  extern __shared__ float sm[];
  float* A = sm; float* Y = sm + MAT; float* Z = sm + 2 * MAT; float* W = sm + 3 * MAT;
  for (int i = threadIdx.x; i < MAT; i += TPB)
    A[i] = 0.5f * (Cb[i] + Cb[((i & 63) << 6) | (i >> 6)]);
  __syncthreads();
  expm_inplace(A, Y, Z, W);
  ew_add_identity(A);           // I + exp(C)
  logm_inplace(A, Y, Z, W);     // B_spd = softplus_m(C)
  sqrtm_inplace(A, Y, Z, W);    // Bh
  for (int i = threadIdx.x; i < MAT; i += TPB) Bh[i] = A[i];
}

// out_b = powm(Bh * (Gih X_b Gih) * Bh, p) = expm(p * logm(Xb)), symmetrized
__global__ __launch_bounds__(TPB) void out_pass_kernel(
    const float* __restrict__ X, const float* __restrict__ Gih,
    const float* __restrict__ Bh, const float* __restrict__ scal,
    float* __restrict__ out) {
  extern __shared__ float sm[];
  float* A = sm; float* Y = sm + MAT; float* Z = sm + 2 * MAT; float* W = sm + 3 * MAT;
  const size_t b = blockIdx.x;
  for (int i = threadIdx.x; i < MAT; i += TPB) { A[i] = X[b * MAT + i]; Y[i] = Gih[i]; }
  gemm64(Y, A, Z);
  gemm64(Z, Y, A);              // Xw = Gih X Gih
  for (int i = threadIdx.x; i < MAT; i += TPB) Y[i] = Bh[i];
  gemm64(Y, A, Z);
  gemm64(Z, Y, A);              // Xb = Bh Xw Bh
  logm_inplace(A, Y, Z, W);
  ew_scale(A, scal[0]);         // p * logm(Xb)
  expm_inplace(A, Y, Z, W);
  for (int i = threadIdx.x; i < MAT; i += TPB)
    out[b * MAT + i] = 0.5f * (A[i] + A[((i & 63) << 6) | (i >> 6)]);
}

// ------------------------------ launcher -----------------------------------
extern "C" void kernel_launch(void* const* d_in, const int* in_sizes, int n_in,
                              void* d_out, int out_size, void* d_ws, size_t ws_size,
                              hipStream_t stream) {
  const float* X       = (const float*)d_in[0];   // [4096,64,64]
  const float* covbias = (const float*)d_in[1];   // [64,64]
  const float* stdbias = (const float*)d_in[2];   // [1]
  float* out = (float*)d_out;

  float* wsf     = (float*)d_ws;
  float* G       = wsf;               // 4096 floats
  float* Gh      = wsf + 4096;
  float* Gih     = wsf + 8192;
  float* Bh      = wsf + 12288;
  float* Tm      = wsf + 16384;
  float* scal    = wsf + 20480;       // 1 float
  float* partial = wsf + 20608;       // 4096 floats
  float* Lbuf    = wsf + 32768;       // 4096*4096 floats (64 MB, lives in L2)

  const size_t SMEM = 4 * MAT * sizeof(float);  // 64 KB LDS / workgroup
  const dim3 redGrid(MAT / TPB);                // 16 blocks
  const dim3 batchGrid(BATCH);

  // G0 = arithmetic mean of X
  mean_batch_kernel<<<redGrid, TPB, 0, stream>>>(X, G);

  // Karcher flow
  for (int it = 0; it < 3; ++it) {
    g_prep_kernel<<<1, TPB, SMEM, stream>>>(G, Gh, Gih);
    karcher_pass_kernel<<<batchGrid, TPB, SMEM, stream>>>(X, Gih, Lbuf);
    mean_batch_kernel<<<redGrid, TPB, 0, stream>>>(Lbuf, Tm);
    g_update_kernel<<<1, TPB, SMEM, stream>>>(Tm, Gh, G);
  }

  // final Gih, scalar std, bias
  g_prep_kernel<<<1, TPB, SMEM, stream>>>(G, Gh, Gih);
  std_pass_kernel<<<batchGrid, TPB, SMEM, stream>>>(X, Gih, partial);
  finalize_kernel<<<1, TPB, 0, stream>>>(partial, stdbias, scal);
  bias_prep_kernel<<<1, TPB, SMEM, stream>>>(covbias, Bh);

  // whiten + bias + matrix power -> output
  out_pass_kernel<<<batchGrid, TPB, SMEM, stream>>>(X, Gih, Bh, scal, out);

  (void)in_sizes; (void)n_in; (void)out_size; (void)ws_size;
}